// DAGrid_28707561407013
// MI455X (gfx1250) — compile-verified
//
#include <hip/hip_runtime.h>
#include <math.h>
#include <stdint.h>

#define NLEV 8
#define EPSF 1e-6f
#define BLK  256
#define FDIM (3 + 6 * NLEV) /* 51 floats per point */

#if __has_builtin(__builtin_amdgcn_global_load_async_to_lds_b32)
#define HAVE_ASYNC_LOAD 1
#else
#define HAVE_ASYNC_LOAD 0
#endif

#if __has_builtin(__builtin_amdgcn_global_store_async_from_lds_b32)
#define HAVE_ASYNC_STORE 1
#else
#define HAVE_ASYNC_STORE 0
#endif

typedef __attribute__((address_space(1))) int gi32_t;  // global int
typedef __attribute__((address_space(3))) int li32_t;  // LDS int

__device__ __forceinline__ void wait_async_zero() {
#if __has_builtin(__builtin_amdgcn_s_wait_asynccnt)
    __builtin_amdgcn_s_wait_asynccnt(0);
#else
    asm volatile("s_wait_asynccnt 0x0" ::: "memory");
#endif
}

// ---------------------------------------------------------------------------
// Kernel 1: transform anchor table -> per-entry {sin(v*2^l) x3, cos(v*2^l) x3}
// Entry's level l is fixed by which [level_offsets[l], level_offsets[l+1])
// range it falls in, so this is point-independent and runs once. Result is
// ~87 MB -> resident in the 192 MB L2 for the gather kernel.
// ---------------------------------------------------------------------------
__global__ __launch_bounds__(BLK) void dagrid_sincos_table(
    const float* __restrict__ data, const int* __restrict__ loffs,
    float* __restrict__ tab, int T)
{
    int t = blockIdx.x * BLK + threadIdx.x;
    if (t >= T) return;
    int l = NLEV - 1;
#pragma unroll
    for (int i = NLEV - 1; i >= 1; --i)
        if (t < loffs[i]) l = i - 1;
    float freq = (float)(1 << l);
    const float* d = data + (size_t)t * 3;
    float v0 = d[0] * freq, v1 = d[1] * freq, v2 = d[2] * freq;
    float* o = tab + (size_t)t * 6;
    o[0] = sinf(v0); o[1] = sinf(v1); o[2] = sinf(v2);
    o[3] = cosf(v0); o[4] = cosf(v1); o[5] = cosf(v2);
}

// ---------------------------------------------------------------------------
// Kernel 2: main trilinear gather + weighted sum.
// USE_TAB=true : tbl = precomputed sin/cos table (6 floats / entry)
// USE_TAB=false: tbl = raw anchor data (3 floats / entry), trig inline
// Full blocks: xyz staged in via async global->LDS; output tile staged in LDS
// and drained via async LDS->global stores (1 cacheline per instruction).
// ---------------------------------------------------------------------------
template <bool USE_TAB>
__global__ __launch_bounds__(BLK) void dagrid_main(
    const float* __restrict__ xyz,
    const float* __restrict__ tbl,
    const float* __restrict__ scales,
    const int*   __restrict__ loffs,
    const float* __restrict__ bounds,
    float* __restrict__ out, int N)
{
    __shared__ float sxyz[BLK * 3];
    __shared__ float tile[BLK * FDIM]; // 52224 B output staging tile

    const int tid  = threadIdx.x;
    const int base = blockIdx.x * BLK;
    const int n    = base + tid;
    const bool full = (base + BLK) <= N; // uniform per block

    // ---- stage this block's xyz tile into LDS (coalesced) ----
#if HAVE_ASYNC_LOAD
    if (full) {
#pragma unroll
        for (int k = 0; k < 3; ++k) {
            const float* src = xyz + (size_t)base * 3 + (size_t)k * BLK + tid;
            __builtin_amdgcn_global_load_async_to_lds_b32(
                (gi32_t*)src, (li32_t*)&sxyz[k * BLK + tid], 0, 0);
        }
        wait_async_zero();     // per-wave ASYNCcnt drain
        __syncthreads();       // cross-wave visibility
    } else
#endif
    {
#pragma unroll
        for (int k = 0; k < 3; ++k) {
            int i  = k * BLK + tid;
            int gi = base * 3 + i;
            sxyz[i] = (gi < N * 3) ? xyz[gi] : 0.0f;
        }
        __syncthreads();
    }

    if (n < N) {
        // gfx1250 prefetch of the next block's xyz tile.
        size_t pf = (size_t)(base + BLK) * 3 + tid;
        if (pf < (size_t)N * 3) __builtin_prefetch(xyz + pf, 0, 1);

        float px = sxyz[tid * 3 + 0];
        float py = sxyz[tid * 3 + 1];
        float pz = sxyz[tid * 3 + 2];

        float lox = bounds[0], loy = bounds[1], loz = bounds[2];
        float dx_ = bounds[3] - lox, dy_ = bounds[4] - loy, dz_ = bounds[5] - loz;
        float size = fmaxf(dx_, fmaxf(dy_, dz_));
        float hix = bounds[3] - EPSF, hiy = bounds[4] - EPSF, hiz = bounds[5] - EPSF;

        float xn0 = (fminf(fmaxf(px, lox), hix) - lox) / size;
        float xn1 = (fminf(fmaxf(py, loy), hiy) - loy) / size;
        float xn2 = (fminf(fmaxf(pz, loz), hiz) - loz) / size;

        // full blocks write to the LDS tile (stride 51 is coprime with 64
        // banks -> conflict free); tail block writes straight to global.
        float* po = full ? &tile[tid * FDIM] : (out + (size_t)n * FDIM);
        po[0] = px; po[1] = py; po[2] = pz;

        for (int l = 0; l < NLEV; ++l) {
            float sc   = scales[l];
            int   loff = loffs[l];
            int   r1   = (int)sc + 1;

            float fx = xn0 * sc, fy = xn1 * sc, fz = xn2 * sc;
            float bfx = floorf(fx), bfy = floorf(fy), bfz = floorf(fz);
            int bx = (int)bfx, by = (int)bfy, bz = (int)bfz;
            float ox = fx - bfx, oy = fy - bfy, oz = fz - bfz;

            int ind000 = (bx * r1 + by) * r1 + bz + loff;

            float wx[2] = {1.0f - ox, ox};
            float wy[2] = {1.0f - oy, oy};
            float wz[2] = {1.0f - oz, oz};

            float a0 = 0.f, a1 = 0.f, a2 = 0.f, a3 = 0.f, a4 = 0.f, a5 = 0.f;
            float freq = (float)(1 << l);

#pragma unroll
            for (int cx = 0; cx < 2; ++cx) {
#pragma unroll
                for (int cy = 0; cy < 2; ++cy) {
                    int indxy = ind000 + cx * r1 * r1 + cy * r1;
                    float wxy = wx[cx] * wy[cy];
#pragma unroll
                    for (int cz = 0; cz < 2; ++cz) {
                        int ind = indxy + cz;
                        float w = wxy * wz[cz];
                        if (USE_TAB) {
                            // 24B entries, 8B-aligned -> three b64 loads (L2 hits)
                            const float2* e = (const float2*)(tbl + (size_t)ind * 6);
                            float2 s01 = e[0], s2c0 = e[1], c12 = e[2];
                            a0 += w * s01.x;  a1 += w * s01.y;  a2 += w * s2c0.x;
                            a3 += w * s2c0.y; a4 += w * c12.x;  a5 += w * c12.y;
                        } else {
                            const float* e = tbl + (size_t)ind * 3;
                            float v0 = e[0] * freq, v1 = e[1] * freq, v2 = e[2] * freq;
                            a0 += w * sinf(v0); a1 += w * sinf(v1); a2 += w * sinf(v2);
                            a3 += w * cosf(v0); a4 += w * cosf(v1); a5 += w * cosf(v2);
                        }
                    }
                }
            }
            float* q = po + 3 + l * 6;
            q[0] = a0; q[1] = a1; q[2] = a2; q[3] = a3; q[4] = a4; q[5] = a5;
        }
    }

    // ---- drain the output tile coalesced: one 128B cacheline per store ----
    if (full) {
        __syncthreads(); // tile fully written by all waves
#if HAVE_ASYNC_STORE
#pragma unroll 1
        for (int k = 0; k < FDIM; ++k) {
            int idx = k * BLK + tid;
            float* gdst = out + (size_t)base * FDIM + idx;
            __builtin_amdgcn_global_store_async_from_lds_b32(
                (gi32_t*)gdst, (li32_t*)&tile[idx], 0, 0);
        }
        wait_async_zero(); // (S_ENDPGM would also drain counters)
#else
#pragma unroll 1
        for (int k = 0; k < FDIM; ++k) {
            int idx = k * BLK + tid;
            out[(size_t)base * FDIM + idx] = tile[idx];
        }
#endif
    }
}

// ---------------------------------------------------------------------------
extern "C" void kernel_launch(void* const* d_in, const int* in_sizes, int n_in,
                              void* d_out, int out_size, void* d_ws, size_t ws_size,
                              hipStream_t stream)
{
    const float* xyz    = (const float*)d_in[0];
    const float* data   = (const float*)d_in[1];
    const float* scales = (const float*)d_in[2];
    const int*   loffs  = (const int*)d_in[3];
    const float* bounds = (const float*)d_in[4];
    float*       out    = (float*)d_out;

    const int N = in_sizes[0] / 3;
    const int T = in_sizes[1] / 3;

    const size_t need = (size_t)T * 6 * sizeof(float); // ~87 MB, L2-resident
    if (ws_size >= need) {
        float* tab = (float*)d_ws;
        dagrid_sincos_table<<<(T + BLK - 1) / BLK, BLK, 0, stream>>>(data, loffs, tab, T);
        dagrid_main<true><<<(N + BLK - 1) / BLK, BLK, 0, stream>>>(
            xyz, tab, scales, loffs, bounds, out, N);
    } else {
        dagrid_main<false><<<(N + BLK - 1) / BLK, BLK, 0, stream>>>(
            xyz, data, scales, loffs, bounds, out, N);
    }
}